// DTWFeatures_14259291422992
// MI455X (gfx1250) — compile-verified
//
#include <hip/hip_runtime.h>
#include <math.h>

typedef __attribute__((ext_vector_type(2))) float v2f;
typedef __attribute__((ext_vector_type(8))) float v8f;

#define B_   32
#define D_   16
#define T_   1024
#define P_   64
#define L_   32
#define CT   16              // time-chunk size
#define NCH  (T_ / CT)       // 64 chunks
#define TSTR (P_ * L_ + 1)   // padded floats per t' row (2049) -> conflict-free reads
#define BIGF 1e30f

#define NWPROD 16            // producer waves
#define TPW    8             // (p, l-half) tiles per producer wave (128 total)
#define NTHREADS ((NWPROD + 2) * 32)   // 576: +2 recurrence waves

#if defined(__has_builtin)
#  if __has_builtin(__builtin_amdgcn_sched_barrier)
#    define SCHED_FENCE() __builtin_amdgcn_sched_barrier(0)
#  endif
#endif
#ifndef SCHED_FENCE
#  define SCHED_FENCE() ((void)0)
#endif

// One block per batch b.
//   waves 0-15 : WMMA producers of cost tiles (cost = x2 + p2 - 2*x.p)
//   waves 16-17: DTW recurrence, lane = pattern p, column of 32 in registers
__global__ __launch_bounds__(NTHREADS, 1)
void dtw_wmma_kernel(const float* __restrict__ x,
                     const float* __restrict__ patts,
                     const float* __restrict__ wp,
                     float* __restrict__ out)
{
    __shared__ float costBuf[2][CT * TSTR];   // double-buffered cost chunk
    __shared__ float p2s[P_ * L_];            // ||patts[p,:,l]||^2

    const int b    = blockIdx.x;
    const int tid  = threadIdx.x;
    const int wave = tid >> 5;
    const int lane = tid & 31;

    const float* xb = x + (size_t)b * D_ * T_;

    if (wave < NWPROD) {
        // -------------------- producer path --------------------
        // p2s: wave w covers p in [w*4, w*4+4), lane = l
        for (int j = 0; j < 4; ++j) {
            int p = wave * 4 + j;
            float s = 0.f;
            for (int d = 0; d < D_; ++d) {
                float v = patts[(p * D_ + d) * L_ + lane];
                s += v * v;
            }
            p2s[p * L_ + lane] = s;
        }

        // A fragments: A[m=l_row, k=d] = -2*patts[p, d, l], 16x4 f32 frag:
        //   lanes 0-15: M rows, VGPR0=K0, VGPR1=K1; lanes 16-31: K2,K3.
        const int lrow_lo = lane & 15;
        const int dbase   = (lane < 16) ? 0 : 2;
        v2f aF[TPW][4];
        #pragma unroll
        for (int i = 0; i < TPW; ++i) {
            int tIdx = wave * TPW + i;   // tile = (p, l-half)
            int p    = tIdx >> 1;
            int lb   = tIdx & 1;
            int lrow = lb * 16 + lrow_lo;
            #pragma unroll
            for (int kk = 0; kk < 4; ++kk) {
                aF[i][kk].x = -2.0f * patts[(p * D_ + 4 * kk + dbase + 0) * L_ + lrow];
                aF[i][kk].y = -2.0f * patts[(p * D_ + 4 * kk + dbase + 1) * L_ + lrow];
            }
        }

        for (int k = 0; k <= NCH; ++k) {
            if (k < NCH) {
                const int t0  = k * CT;
                float* buf    = costBuf[k & 1];
                const int tt  = t0 + (lane & 15);

                // B fragment: B[k=d, n=t'] = x[b, d, t0+t'] (4x16 f32 frag, mirror of A)
                v2f bx[4];
                float x2p = 0.f;
                #pragma unroll
                for (int kk = 0; kk < 4; ++kk) {
                    float e0 = xb[(4 * kk + dbase + 0) * T_ + tt];
                    float e1 = xb[(4 * kk + dbase + 1) * T_ + tt];
                    bx[kk].x = e0; bx[kk].y = e1;
                    x2p += e0 * e0 + e1 * e1;
                }
                // full x2[t'] = partial(lane) + partial(lane^16)
                float x2lane = x2p + __shfl_xor(x2p, 16, 32);

                const int ladd = ((lane >= 16) ? 8 : 0);
                const int tq   = lane & 15;
                #pragma unroll
                for (int i = 0; i < TPW; ++i) {
                    int tIdx = wave * TPW + i;
                    int p    = tIdx >> 1;
                    int lb   = tIdx & 1;
                    v8f c = {0.f, 0.f, 0.f, 0.f, 0.f, 0.f, 0.f, 0.f};
                    #pragma unroll
                    for (int kk = 0; kk < 4; ++kk) {
                        c = __builtin_amdgcn_wmma_f32_16x16x4_f32(
                                false, aF[i][kk], false, bx[kk],
                                (short)0, c, false, false);
                    }
                    // C layout: VGPR v -> M = v (+8 for lanes>=16), N = lane&15
                    #pragma unroll
                    for (int v = 0; v < 8; ++v) {
                        int l = lb * 16 + ladd + v;
                        float val = c[v] + p2s[p * L_ + l] + x2lane;
                        buf[tq * TSTR + l * P_ + p] = val;
                    }
                    // limit scheduler tile interleaving -> fewer live C accs,
                    // keeps all A-fragments resident (no scratch spills)
                    SCHED_FENCE();
                }
            }
            __syncthreads();
        }
    } else {
        // -------------------- recurrence path --------------------
        const int p    = (wave - NWPROD) * 32 + lane;   // 64 chains
        const float w  = *wp;
        float* outp    = out + ((size_t)b * P_ + p) * T_;
        float prev[L_];
        #pragma unroll
        for (int l = 0; l < L_; ++l) prev[l] = 0.f;

        for (int k = 0; k <= NCH; ++k) {
            if (k > 0) {
                const int kc      = k - 1;
                const float* buf  = costBuf[kc & 1];
                const int t0      = kc * CT;
                for (int tq = 0; tq < CT; ++tq) {
                    const int t       = t0 + tq;
                    const float* row  = buf + tq * TSTR + p;
                    if (t == 0) {
                        // first time column: cumulative sum along l
                        float run = 0.f;
                        #pragma unroll
                        for (int l = 0; l < L_; ++l) {
                            run += row[l * P_];
                            prev[l] = run;
                        }
                    } else {
                        // D[l,t] = cost + min(D[l-1,t], w*min(D[l,t-1], D[l-1,t-1]))
                        float carry = BIGF;   // new_{l-1}
                        float pm1   = BIGF;   // prev[l-1] (old)
                        #pragma unroll
                        for (int l = 0; l < L_; ++l) {
                            float pl = prev[l];
                            float mm = w * fminf(pl, pm1);
                            float nv = row[l * P_] + fminf(carry, mm);
                            prev[l] = nv;
                            carry = nv;
                            pm1 = pl;
                        }
                    }
                    outp[t] = sqrtf(prev[L_ - 1]);
                }
            }
            __syncthreads();
        }
    }
}

extern "C" void kernel_launch(void* const* d_in, const int* in_sizes, int n_in,
                              void* d_out, int out_size, void* d_ws, size_t ws_size,
                              hipStream_t stream) {
    (void)in_sizes; (void)n_in; (void)d_ws; (void)ws_size; (void)out_size;
    const float* x     = (const float*)d_in[0];
    const float* patts = (const float*)d_in[1];
    const float* w     = (const float*)d_in[2];
    float* out         = (float*)d_out;
    dtw_wmma_kernel<<<dim3(B_), dim3(NTHREADS), 0, stream>>>(x, patts, w, out);
}